// MSCAFusion_21148418965567
// MI455X (gfx1250) — compile-verified
//
#include <hip/hip_runtime.h>

// ---------------------------------------------------------------------------
// MSCA fusion for MI455X (gfx1250, wave32).  All GEMMs via V_WMMA_F32_16X16X4_F32
// (pure f32 WMMA) to preserve top-k boundary fidelity.  Two softmax branches are
// fused into one combined-weight matrix (top-341 support is a subset of top-512).
// ---------------------------------------------------------------------------

typedef float v2f __attribute__((ext_vector_type(2)));
typedef float v8f __attribute__((ext_vector_type(8)));

// D = A(16x4) * B(4x16) + C, f32, one wave.
__device__ __forceinline__ v8f wmma_f32_16x16x4(v2f a, v2f b, v8f c) {
#if __has_builtin(__builtin_amdgcn_wmma_f32_16x16x4_f32)
  return __builtin_amdgcn_wmma_f32_16x16x4_f32(false, a, false, b, (short)0, c,
                                               false, false);
#else
  // Correct shuffle-based emulation of the ISA data layout (slow fallback).
  int lane = threadIdx.x & 31;
  int hf = lane >> 4, l16 = lane & 15;
  #pragma unroll
  for (int r = 0; r < 8; ++r) {
    int m = r + hf * 8, n = l16;
    float acc = 0.f;
    #pragma unroll
    for (int k = 0; k < 4; ++k) {
      float av = __shfl((k & 1) ? a.y : a.x, m + ((k >= 2) ? 16 : 0), 32);
      float bv = __shfl((k & 1) ? b.y : b.x, n + ((k >= 2) ? 16 : 0), 32);
      acc += av * bv;
    }
    c[r] += acc;
  }
  return c;
#endif
}

__device__ __forceinline__ float wred_sum(float v) {
  for (int o = 16; o; o >>= 1) v += __shfl_xor(v, o, 32);
  return v;
}
__device__ __forceinline__ int wred_sumi(int v) {
  for (int o = 16; o; o >>= 1) v += __shfl_xor(v, o, 32);
  return v;
}
__device__ __forceinline__ float wred_max(float v) {
  for (int o = 16; o; o >>= 1) v = fmaxf(v, __shfl_xor(v, o, 32));
  return v;
}
// Order-preserving float->uint key (larger float <=> larger unsigned).
__device__ __forceinline__ unsigned fkey(float f) {
  unsigned u = __float_as_uint(f);
  return (u & 0x80000000u) ? ~u : (u ^ 0x80000000u);
}

// Wave-cooperative MSB-first radix select: key of the k-th largest of the
// 1024 keys held 32-per-lane.  All lanes return the same value.
__device__ __forceinline__ unsigned radix_kth(const unsigned keys[32], int k) {
  unsigned prefix = 0u, maskAcc = 0u;
  for (int bit = 31; bit >= 0; --bit) {
    unsigned bm = 1u << bit;
    unsigned tst = prefix | bm;
    unsigned nm = maskAcc | bm;
    int c = 0;
    #pragma unroll
    for (int i = 0; i < 32; ++i) c += (int)((keys[i] & nm) == tst);
    c = wred_sumi(c);
    if (c >= k) prefix = tst; else k -= c;
    maskAcc = nm;
  }
  return prefix;
}

// D(16x16) = A(16xK) * W^T : A row-major [16][lda], W row-major [16][ldw],
// contraction over contiguous K of both.
__device__ __forceinline__ v8f gemm_tile_AtWt(const float* __restrict__ A, int lda,
                                              const float* __restrict__ W, int ldw,
                                              int K) {
  int lane = threadIdx.x & 31;
  int hf = lane >> 4, l16 = lane & 15;
  const float* arow = A + l16 * lda;
  const float* wrow = W + l16 * ldw;
  v8f acc = {};
  for (int kk = 0; kk < K; kk += 4) {
    v2f a = *(const v2f*)(arow + kk + 2 * hf);
    v2f b = *(const v2f*)(wrow + kk + 2 * hf);
    acc = wmma_f32_16x16x4(a, b, acc);
  }
  return acc;
}

// D(16x16) = A(16xK) * B(Kx16) : B row-major [K][ldb].
__device__ __forceinline__ v8f gemm_tile_A_Brm(const float* __restrict__ A, int lda,
                                               const float* __restrict__ Bm, int ldb,
                                               int K) {
  int lane = threadIdx.x & 31;
  int hf = lane >> 4, l16 = lane & 15;
  const float* arow = A + l16 * lda;
  v8f acc = {};
  for (int kk = 0; kk < K; kk += 4) {
    v2f a = *(const v2f*)(arow + kk + 2 * hf);
    v2f b;
    b.x = Bm[(kk + 2 * hf) * ldb + l16];
    b.y = Bm[(kk + 2 * hf + 1) * ldb + l16];
    acc = wmma_f32_16x16x4(a, b, acc);
  }
  return acc;
}

// ---------------- Kernel 1: transpose x [B,C,N] -> xq [B*N, C] --------------
__global__ __launch_bounds__(256) void k_transpose_x(const float* __restrict__ x,
                                                     float* __restrict__ xq) {
  int i = blockIdx.x * 256 + threadIdx.x;          // 4*256*1024 elements
  int n = i & 1023;
  int c = (i >> 10) & 255;
  int b = i >> 18;
  xq[((b << 10) + n) * 256 + c] = x[i];
}

// ------- Kernel 2: multi-scale avgpool (3/5/7, pad, incl-pad) -> [B*N, C] ----
__global__ __launch_bounds__(256) void k_pool(const float* __restrict__ y,
                                              float* __restrict__ yk) {
  int i = blockIdx.x * 256 + threadIdx.x;
  int w = i & 31;
  int h = (i >> 5) & 31;
  int c = (i >> 10) & 255;
  int b = i >> 18;
  const float* plane = y + ((b * 256 + c) << 10);
  float acc = 0.f;
  #pragma unroll
  for (int dy = -3; dy <= 3; ++dy) {
    int hh = h + dy;
    if (hh < 0 || hh > 31) continue;
    #pragma unroll
    for (int dx = -3; dx <= 3; ++dx) {
      int ww = w + dx;
      if (ww < 0 || ww > 31) continue;
      int ady = dy < 0 ? -dy : dy, adx = dx < 0 ? -dx : dx;
      float wt = 1.f / 49.f;
      if (ady <= 2 && adx <= 2) wt += 1.f / 25.f;
      if (ady <= 1 && adx <= 1) wt += 1.f / 9.f;
      acc += wt * plane[hh * 32 + ww];
    }
  }
  yk[((b << 10) + h * 32 + w) * 256 + c] = acc;
}

// ------------- Kernel 3: LayerNorm over C (in place on yk) ------------------
__global__ __launch_bounds__(256) void k_ln(float* __restrict__ yk,
                                            const float* __restrict__ g,
                                            const float* __restrict__ bn) {
  __shared__ float red[256];
  int row = blockIdx.x;
  int tid = threadIdx.x;
  float z = yk[row * 256 + tid];
  red[tid] = z;
  __syncthreads();
  for (int s = 128; s; s >>= 1) {
    if (tid < s) red[tid] += red[tid + s];
    __syncthreads();
  }
  float mu = red[0] * (1.f / 256.f);
  __syncthreads();
  float dz = z - mu;
  red[tid] = dz * dz;
  __syncthreads();
  for (int s = 128; s; s >>= 1) {
    if (tid < s) red[tid] += red[tid + s];
    __syncthreads();
  }
  float var = red[0] * (1.f / 256.f);
  yk[row * 256 + tid] = dz * rsqrtf(var + 1e-5f) * g[tid] + bn[tid];
}

// ------------- Kernel 4: q = xq @ Wq^T -> [B,NH,N,hd] ------------------------
__global__ __launch_bounds__(256) void k_qproj(const float* __restrict__ A,
                                               const float* __restrict__ W,
                                               float* __restrict__ q) {
  int wave = threadIdx.x >> 5;
  int tile = blockIdx.x * 8 + wave;                // 4096 tiles: 256 M x 16 P
  int pt = tile & 15, mt = tile >> 4;
  v8f d = gemm_tile_AtWt(A + mt * 16 * 256, 256, W + pt * 16 * 256, 256, 256);
  int lane = threadIdx.x & 31, hf = lane >> 4, l16 = lane & 15;
  int p = pt * 16 + l16;
  int h = p >> 5, dd = p & 31;
  #pragma unroll
  for (int r = 0; r < 8; ++r) {
    int m = mt * 16 + r + hf * 8;
    int b = m >> 10, n = m & 1023;
    q[((b * 8 + h) * 1024 + n) * 32 + dd] = d[r];
  }
}

// ------------- Kernel 5: kv = yk @ Wkv^T -> k [B,NH,N,hd], v [B,NH,N,hd] ----
__global__ __launch_bounds__(256) void k_kvproj(const float* __restrict__ A,
                                                const float* __restrict__ W,
                                                float* __restrict__ kb,
                                                float* __restrict__ vb) {
  int wave = threadIdx.x >> 5;
  int tile = blockIdx.x * 8 + wave;                // 8192 tiles: 256 M x 32 P
  int pt = tile & 31, mt = tile >> 5;
  v8f d = gemm_tile_AtWt(A + mt * 16 * 256, 256, W + pt * 16 * 256, 256, 256);
  int lane = threadIdx.x & 31, hf = lane >> 4, l16 = lane & 15;
  int p = pt * 16 + l16;
  #pragma unroll
  for (int r = 0; r < 8; ++r) {
    int m = mt * 16 + r + hf * 8;
    int b = m >> 10, n = m & 1023;
    if (p < 256) {
      int h = p >> 5, dd = p & 31;
      kb[((b * 8 + h) * 1024 + n) * 32 + dd] = d[r];
    } else {
      int p2 = p - 256;
      int h = p2 >> 5, dd = p2 & 31;
      vb[((b * 8 + h) * 1024 + n) * 32 + dd] = d[r];
    }
  }
}

// ------------- Kernel 6: dual-top-k attention, fused ------------------------
// One workgroup = (b, h, 16 query rows).  LDS holds the 16x1024 score tile.
__global__ __launch_bounds__(256) void k_attn(const float* __restrict__ q,
                                              const float* __restrict__ kb,
                                              const float* __restrict__ vb,
                                              const float* __restrict__ aw1p,
                                              const float* __restrict__ aw2p,
                                              float* __restrict__ o) {
  __shared__ float S[16 * 1024];                   // 64 KB
  int bid = blockIdx.x;
  int nblk = bid & 63;
  int bh = bid >> 6;
  int h = bh & 7, b = bh >> 3;
  int n0 = nblk * 16;
  int wave = threadIdx.x >> 5;
  int lane = threadIdx.x & 31;
  int hf = lane >> 4, l16 = lane & 15;
  const float* Q = q + (bh * 1024 + n0) * 32;
  const float* K = kb + bh * 1024 * 32;
  const float* V = vb + bh * 1024 * 32;
  const float scale = 0.17677669529663687f;        // 1/sqrt(32)

  // --- S = scale * Q @ K^T ; wave handles 8 column tiles -------------------
  for (int j = 0; j < 8; ++j) {
    int c0 = (wave * 8 + j) * 16;
    v8f d = gemm_tile_AtWt(Q, 32, K + c0 * 32, 32, 32);
    #pragma unroll
    for (int r = 0; r < 8; ++r)
      S[(r + hf * 8) * 1024 + c0 + l16] = d[r] * scale;
  }
  __syncthreads();

  float w1 = aw1p[0], w2 = aw2p[0];

  // --- per-row exact top-k thresholds + fused dual softmax -> weights ------
  for (int rr = 0; rr < 2; ++rr) {
    int row = wave * 2 + rr;
    float vals[32];
    unsigned keys[32];
    float mx = -3.402823466e38f;
    #pragma unroll
    for (int i = 0; i < 32; ++i) {
      float v = S[row * 1024 + lane + 32 * i];
      vals[i] = v;
      keys[i] = fkey(v);
      mx = fmaxf(mx, v);
    }
    mx = wred_max(mx);
    unsigned kt1 = radix_kth(keys, 512);           // N1 // 2
    unsigned kt2 = radix_kth(keys, 341);           // N1 // 3
    float s1 = 0.f, s2 = 0.f;
    #pragma unroll
    for (int i = 0; i < 32; ++i) {
      float e = __expf(vals[i] - mx);
      if (keys[i] >= kt1) s1 += e;
      if (keys[i] >= kt2) s2 += e;
    }
    s1 = wred_sum(s1);
    s2 = wred_sum(s2);
    float r1 = w1 / s1, r2 = w2 / s2;
    #pragma unroll
    for (int i = 0; i < 32; ++i) {
      float e = __expf(vals[i] - mx);
      float cw = 0.f;
      if (keys[i] >= kt1) cw += r1 * e;
      if (keys[i] >= kt2) cw += r2 * e;
      S[row * 1024 + lane + 32 * i] = cw;
    }
  }
  __syncthreads();

  // --- O = CW(16x1024) @ V(1024x32) ; split: 2 d-tiles x 4 K-quarters ------
  int tc = wave & 1, kq = wave >> 1;
  v8f acc = gemm_tile_A_Brm(S + kq * 256, 1024, V + kq * 256 * 32 + tc * 16, 32, 256);
  __syncthreads();                                  // all CW reads done
  #pragma unroll
  for (int r = 0; r < 8; ++r)
    S[wave * 256 + r * 32 + lane] = acc[r];         // partials, 8x256 floats
  __syncthreads();

  int tid = threadIdx.x;
  #pragma unroll
  for (int t2 = 0; t2 < 2; ++t2) {
    float sum = 0.f;
    #pragma unroll
    for (int kq2 = 0; kq2 < 4; ++kq2) sum += S[(kq2 * 2 + t2) * 256 + tid];
    int r = tid >> 5;
    int l = tid & 31;
    int m = r + ((l & 16) ? 8 : 0);
    int dcol = t2 * 16 + (l & 15);
    o[(b * 1024 + n0 + m) * 256 + h * 32 + dcol] = sum;
  }
}

// ------------- Kernel 7: out = o @ Wproj^T + b, store [B,C,H,W] -------------
__global__ __launch_bounds__(256) void k_outproj(const float* __restrict__ A,
                                                 const float* __restrict__ W,
                                                 const float* __restrict__ bias,
                                                 float* __restrict__ out) {
  int wave = threadIdx.x >> 5;
  int tile = blockIdx.x * 8 + wave;                // 4096 tiles: 256 M x 16 P
  int pt = tile & 15, mt = tile >> 4;
  v8f d = gemm_tile_AtWt(A + mt * 16 * 256, 256, W + pt * 16 * 256, 256, 256);
  int lane = threadIdx.x & 31, hf = lane >> 4, l16 = lane & 15;
  int p = pt * 16 + l16;
  float bv = bias[p];
  #pragma unroll
  for (int r = 0; r < 8; ++r) {
    int m = mt * 16 + r + hf * 8;
    int b = m >> 10, n = m & 1023;
    out[(b * 256 + p) * 1024 + n] = d[r] + bv;
  }
}

// ---------------------------------------------------------------------------
extern "C" void kernel_launch(void* const* d_in, const int* in_sizes, int n_in,
                              void* d_out, int out_size, void* d_ws, size_t ws_size,
                              hipStream_t stream) {
  (void)in_sizes; (void)n_in; (void)out_size; (void)ws_size;
  const float* x     = (const float*)d_in[0];
  const float* y     = (const float*)d_in[1];
  const float* Wq    = (const float*)d_in[2];
  const float* Wkv   = (const float*)d_in[3];
  const float* Wproj = (const float*)d_in[4];
  const float* bproj = (const float*)d_in[5];
  const float* ln_g  = (const float*)d_in[6];
  const float* ln_b  = (const float*)d_in[7];
  const float* aw1   = (const float*)d_in[8];
  const float* aw2   = (const float*)d_in[9];
  float* out = (float*)d_out;

  const size_t M = 4096ull * 256ull;               // 1M floats per buffer
  float* ws = (float*)d_ws;
  float* xq = ws + 0 * M;
  float* yk = ws + 1 * M;
  float* q  = ws + 2 * M;
  float* kb = ws + 3 * M;
  float* vb = ws + 4 * M;
  float* o  = ws + 5 * M;

  k_transpose_x<<<4096, 256, 0, stream>>>(x, xq);
  k_pool<<<4096, 256, 0, stream>>>(y, yk);
  k_ln<<<4096, 256, 0, stream>>>(yk, ln_g, ln_b);
  k_qproj<<<512, 256, 0, stream>>>(xq, Wq, q);
  k_kvproj<<<1024, 256, 0, stream>>>(yk, Wkv, kb, vb);
  k_attn<<<2048, 256, 0, stream>>>(q, kb, vb, aw1, aw2, o);
  k_outproj<<<512, 256, 0, stream>>>(o, Wproj, bproj, out);
}